// UDCTModule_2705829397126
// MI455X (gfx1250) — compile-verified
//
#include <hip/hip_runtime.h>
#include <math.h>

// =====================================================================
// UDCT (curvelet-style) transform for MI455X / gfx1250.
//  fft2(image) -> per-band window * fold -> ifft2 -> real part.
// 1D FFTs: n = 16*m split; DFT-16 stage on V_WMMA_F32_16X16X4_F32
// (f32 matrix pipe, keeps reference precision), remaining m-point FFT
// as radix-2 Stockham in LDS. Column passes batch 16 columns per
// workgroup using CDNA5's 320KB LDS for coalesced global access.
// Data movement: TDM tensor_load_to_lds for contiguous row copies
// (TENSORcnt), global_load_async_to_lds_b32 for the transposing
// column staging (ASYNCcnt) -- both bypass the VGPR round-trip.
// =====================================================================

#define IMG_N 1024
#define OUT_TOTAL 28770304

typedef __attribute__((ext_vector_type(2))) float v2f;
typedef __attribute__((ext_vector_type(8))) float v8f;
typedef __attribute__((ext_vector_type(4))) unsigned int v4u;
typedef __attribute__((ext_vector_type(4))) int v4i;
typedef __attribute__((ext_vector_type(8))) int v8i;

__device__ __forceinline__ v8f wmma_f32_16x16x4(v2f a, v2f b, v8f c) {
    // 8-arg form: (neg_a, A, neg_b, B, c_mod, C, reuse_a, reuse_b)
    // f32 WMMA: NEG only applies to C per ISA, so neg_a/neg_b stay false.
    return __builtin_amdgcn_wmma_f32_16x16x4_f32(
        false, a, false, b, (short)0, c, false, false);
}

// ---- TDM: DMA one contiguous row of n floats from global to LDS ----
// D# bitfields per CDNA5 ISA 8.3-8.6. Kernels using this have no static
// __shared__, so dynamic-LDS byte offsets start at 0.
#if __has_builtin(__builtin_amdgcn_tensor_load_to_lds)
#define HAVE_TDM 1
__device__ __forceinline__ void tdm_load_row(unsigned int ldsByteOff,
                                             const void* src, int n)
{
    unsigned long long ga = (unsigned long long)src;
    v4u g0;
    g0[0] = 1u;                                   // count=1 user descriptor
    g0[1] = ldsByteOff;                           // lds_addr
    g0[2] = (unsigned int)ga;                     // global_addr[31:0]
    g0[3] = (unsigned int)(ga >> 32) | (2u << 30);// global_addr[56:32] | type=2
    v8i g1;
    g1[0] = (int)(2u << 16);                      // data_size=4B, mask=0
    g1[1] = (int)((unsigned)(n & 0xFFFF) << 16);  // tensor_dim0[15:0]
    g1[2] = (int)(((unsigned)n >> 16) | (1u << 16)); // tensor_dim0 hi | tensor_dim1=1
    g1[3] = (int)((unsigned)(n & 0xFFFF) << 16);  // tile_dim0 = n
    g1[4] = 1;                                    // tile_dim1 = 1
    g1[5] = n;                                    // tensor_dim0_stride
    g1[6] = 0;
    g1[7] = 0;
    v4i gz = {0, 0, 0, 0};
#if defined(__clang_major__) && (__clang_major__ >= 23)
    v8i gz8 = {0, 0, 0, 0, 0, 0, 0, 0};
    __builtin_amdgcn_tensor_load_to_lds(g0, g1, gz, gz, gz8, 0);
#else
    __builtin_amdgcn_tensor_load_to_lds(g0, g1, gz, gz, 0);
#endif
}
__device__ __forceinline__ void tdm_wait() {
    __builtin_amdgcn_s_wait_tensorcnt(0);
}
#else
#define HAVE_TDM 0
#endif

// ---- Async global->LDS scatter copy (per-lane LDS address) ----
__device__ __forceinline__ void async_ld_b32(unsigned int ldsByteOff,
                                             const void* gaddr)
{
    asm volatile("global_load_async_to_lds_b32 %0, %1, off"
                 :: "v"(ldsByteOff), "v"((unsigned long long)gaddr)
                 : "memory");
}
__device__ __forceinline__ void async_wait() {
    asm volatile("s_wait_asynccnt 0x0" ::: "memory");
}

struct FFTRes { float* re; float* im; };

// In-LDS batched 1D FFT of length n = 16*m over nInst instances.
// x*: input (instance i at x[i*stride + j]), y*: scratch (same layout).
// dir = -1 forward, +1 inverse. Caller must have filled x before call.
// Result: X[k1 + 16*k2] at res[ i*stride + (k&15)*m + (k>>4) ].
__device__ FFTRes fft_core(float* xr, float* xi, float* yr, float* yi,
                           int n, int nInst, int stride, float dir)
{
    const int m = n >> 4;
    const int tid = threadIdx.x;
    const int lane = tid & 31;
    const int wave = tid >> 5;
    const int nwaves = blockDim.x >> 5;
    const int row = lane & 15;   // M (A/C row) or N (B/C col) within tile
    const int hi = lane >> 4;    // lane half selector
    const float TWO_PI = 6.283185307179586f;

    __syncthreads();  // x populated

    // ---- Stage 1: DFT-16 along j1 via WMMA, x viewed as 16 x m ----
    // X[k1+16k2] = sum_j2 Wm^{j2 k2} * T[j2,k1] * sum_j1 W16^{j1 k1} x[j1*m+j2]
    const int tilesPerInst = m >> 4;
    const int nTiles = nInst * tilesPerInst;
    for (int T = wave; T < nTiles; T += nwaves) {      // wave-uniform: EXEC all-1s
        int inst = T / tilesPerInst;
        int tt = T - inst * tilesPerInst;
        int base = inst * stride;
        int j2 = (tt << 4) + row;                      // this lane's column
        v8f cr = {0.f,0.f,0.f,0.f,0.f,0.f,0.f,0.f};
        v8f ci = {0.f,0.f,0.f,0.f,0.f,0.f,0.f,0.f};
        for (int c = 0; c < 4; ++c) {                  // K chunks of 4
            int k0 = (c << 2) + (hi << 1);             // A/B K pair base
            float s0, c0, s1, c1;
            __sincosf(dir * TWO_PI * (float)(row * k0)     * 0.0625f, &s0, &c0);
            __sincosf(dir * TWO_PI * (float)(row * (k0+1)) * 0.0625f, &s1, &c1);
            v2f aRe  = { c0,  c1};                     // Re D16[row][k0..k0+1]
            v2f aIm  = { s0,  s1};                     // Im D16 (dir baked in)
            v2f aImN = {-s0, -s1};
            v2f bRe = { xr[base + k0*m + j2], xr[base + (k0+1)*m + j2] };
            v2f bIm = { xi[base + k0*m + j2], xi[base + (k0+1)*m + j2] };
            cr = wmma_f32_16x16x4(aRe,  bRe, cr);      // Re += Dr*Xr
            cr = wmma_f32_16x16x4(aImN, bIm, cr);      // Re -= Di*Xi
            ci = wmma_f32_16x16x4(aIm,  bRe, ci);      // Im += Di*Xr
            ci = wmma_f32_16x16x4(aRe,  bIm, ci);      // Im += Dr*Xi
        }
        // Twiddle T[j2,k1] = exp(dir*i*2pi*j2*k1/n); write Z to y region.
        for (int r = 0; r < 8; ++r) {
            int k1 = r + (hi << 3);                    // C row per VGPR layout
            float s, cc;
            __sincosf(dir * TWO_PI * (float)(j2 * k1) / (float)n, &s, &cc);
            float zr = cr[r]*cc - ci[r]*s;
            float zi = cr[r]*s  + ci[r]*cc;
            yr[base + k1*m + j2] = zr;
            yi[base + k1*m + j2] = zi;
        }
    }
    __syncthreads();

    // ---- Stage 2: m-point radix-2 Stockham along j2, 16*nInst batches ----
    float* sr = yr; float* si = yi; float* dr = xr; float* di = xi;
    int nn = m, s = 1;
    const int hb = m >> 1;                // butterflies per batch per stage
    const int total = nInst * 16 * hb;
    while (nn > 1) {
        int half = nn >> 1;
        for (int t = tid; t < total; t += blockDim.x) {
            int b = t / hb;
            int u = t - b * hb;
            int p = u / s;
            int q = u - p * s;
            int base = (b >> 4) * stride + (b & 15) * m;
            float wsn, wcs;
            __sincosf(dir * TWO_PI * (float)p / (float)nn, &wsn, &wcs);
            int ia = base + q + s * p;
            int ib = base + q + s * (p + half);
            float ar = sr[ia], ai = si[ia];
            float br = sr[ib], bi = si[ib];
            int oa = base + q + s * (2 * p);
            dr[oa]     = ar + br;
            di[oa]     = ai + bi;
            float fr = ar - br, fi = ai - bi;
            dr[oa + s] = fr * wcs - fi * wsn;
            di[oa + s] = fr * wsn + fi * wcs;
        }
        __syncthreads();
        float* tp;
        tp = sr; sr = dr; dr = tp;
        tp = si; si = di; di = tp;
        nn = half; s <<= 1;
    }
    FFTRes res; res.re = sr; res.im = si;
    return res;
}

// ---------------- Forward: row FFT (real input) ----------------
__global__ void k_fwd_rows(const float* __restrict__ img,
                           float* __restrict__ outr, float* __restrict__ outi)
{
    extern __shared__ float smem[];
    const int n = IMG_N, m = n >> 4;
    float* xr = smem; float* xi = xr + n; float* yr = xi + n; float* yi = yr + n;
    const int r = blockIdx.x;
#if HAVE_TDM
    if (threadIdx.x < 32) {                      // one wave issues the DMA
        tdm_load_row(0u, img + (size_t)r * IMG_N, n);
    }
    for (int j = threadIdx.x; j < n; j += blockDim.x) xi[j] = 0.0f;
    if (threadIdx.x < 32) tdm_wait();
#else
    for (int j = threadIdx.x; j < n; j += blockDim.x) {
        xr[j] = img[(size_t)r * IMG_N + j];
        xi[j] = 0.0f;
    }
#endif
    FFTRes R = fft_core(xr, xi, yr, yi, n, 1, n, -1.0f);
    for (int k = threadIdx.x; k < n; k += blockDim.x) {
        int idx = (k & 15) * m + (k >> 4);
        outr[(size_t)r * IMG_N + k] = R.re[idx];
        outi[(size_t)r * IMG_N + k] = R.im[idx];
    }
}

// ---------------- Forward: column FFT, 16 columns/block ----------------
__global__ void k_fwd_cols(const float* __restrict__ inr, const float* __restrict__ ini,
                           float* __restrict__ outr, float* __restrict__ outi)
{
    extern __shared__ float smem[];
    const int n = IMG_N, m = n >> 4, nInst = 16, stride = n + 4;  // pad: no LDS bank conflicts
    float* xr = smem;
    float* xi = xr + nInst * stride;
    float* yr = xi + nInst * stride;
    float* yi = yr + nInst * stride;
    const unsigned int xrOff = 0u;
    const unsigned int xiOff = (unsigned int)(nInst * stride) * 4u;
    const int c0 = blockIdx.x << 4;
    // Transposing copy: global row-segments -> strided LDS; per-lane LDS
    // addressing via async-load-to-LDS (no VGPR round trip).
    for (int idx = threadIdx.x; idx < nInst * n; idx += blockDim.x) {
        int ci = idx & 15, u = idx >> 4;                 // 64B coalesced segments
        unsigned int l = (unsigned int)(ci * stride + u) * 4u;
        async_ld_b32(xrOff + l, inr + (size_t)u * IMG_N + c0 + ci);
        async_ld_b32(xiOff + l, ini + (size_t)u * IMG_N + c0 + ci);
    }
    async_wait();
    FFTRes R = fft_core(xr, xi, yr, yi, n, nInst, stride, -1.0f);
    for (int idx = threadIdx.x; idx < nInst * n; idx += blockDim.x) {
        int ci = idx & 15, k = idx >> 4;
        int src = ci * stride + (k & 15) * m + (k >> 4);
        outr[(size_t)k * IMG_N + c0 + ci] = R.re[src];
        outi[(size_t)k * IMG_N + c0 + ci] = R.im[src];
    }
}

// ------- Inverse columns, fused window*F + d x d spectrum fold -------
__global__ void k_band_cols(const float* __restrict__ fr, const float* __restrict__ fi,
                            const float* __restrict__ win,       // offset to group base
                            float* __restrict__ tr, float* __restrict__ ti, // group base
                            int nf, int d)
{
    extern __shared__ float smem[];
    const int nInst = 16, stride = nf + 4, m = nf >> 4;
    float* xr = smem;
    float* xi = xr + nInst * stride;
    float* yr = xi + nInst * stride;
    float* yi = yr + nInst * stride;
    const int band = blockIdx.y;
    const float* w = win + (size_t)band * IMG_N * IMG_N;
    float* trb = tr + (size_t)band * nf * nf;
    float* tib = ti + (size_t)band * nf * nf;
    const int c0 = blockIdx.x << 4;
    for (int idx = threadIdx.x; idx < nInst * nf; idx += blockDim.x) {
        int ci = idx & 15, u = idx >> 4;
        float ar = 0.0f, ai = 0.0f;
        for (int p = 0; p < d; ++p) {
            for (int q = 0; q < d; ++q) {
                size_t g = (size_t)(u + p * nf) * IMG_N + (c0 + ci + q * nf);
                float wv = w[g];
                ar += wv * fr[g];     // F resident in 192MB L2 across bands
                ai += wv * fi[g];
            }
        }
        xr[ci * stride + u] = ar;
        xi[ci * stride + u] = ai;
    }
    FFTRes R = fft_core(xr, xi, yr, yi, nf, nInst, stride, +1.0f);
    for (int idx = threadIdx.x; idx < nInst * nf; idx += blockDim.x) {
        int ci = idx & 15, k = idx >> 4;
        int src = ci * stride + (k & 15) * m + (k >> 4);
        trb[(size_t)k * nf + c0 + ci] = R.re[src];
        tib[(size_t)k * nf + c0 + ci] = R.im[src];
    }
}

// ------- Inverse rows, write real part with all scaling folded -------
__global__ void k_band_rows(const float* __restrict__ tr, const float* __restrict__ ti,
                            float* __restrict__ out, int nf, float scale)
{
    extern __shared__ float smem[];
    const int m = nf >> 4;
    float* xr = smem; float* xi = xr + nf; float* yr = xi + nf; float* yi = yr + nf;
    const int band = blockIdx.y, r = blockIdx.x;
    const float* trb = tr + (size_t)band * nf * nf + (size_t)r * nf;
    const float* tib = ti + (size_t)band * nf * nf + (size_t)r * nf;
    float* ob = out + (size_t)band * nf * nf + (size_t)r * nf;
#if HAVE_TDM
    if (threadIdx.x < 32) {                      // one wave drives the TDM
        tdm_load_row(0u, trb, nf);               // -> xr (LDS byte 0)
        tdm_load_row((unsigned int)nf * 4u, tib, nf); // -> xi
        tdm_wait();
    }
#else
    for (int j = threadIdx.x; j < nf; j += blockDim.x) {
        xr[j] = trb[j];
        xi[j] = tib[j];
    }
#endif
    FFTRes R = fft_core(xr, xi, yr, yi, nf, 1, nf, +1.0f);
    for (int k = threadIdx.x; k < nf; k += blockDim.x) {
        ob[k] = R.re[(k & 15) * m + (k >> 4)] * scale;
    }
}

extern "C" void kernel_launch(void* const* d_in, const int* in_sizes, int n_in,
                              void* d_out, int out_size, void* d_ws, size_t ws_size,
                              hipStream_t stream)
{
    const float* image   = (const float*)d_in[0];
    const float* windows = (const float*)d_in[1];
    float* out = (float*)d_out;

    // Workspace: Fr | Fi | Tr | Ti   (Tr/Ti sized for the full output shape;
    // also reused as the forward row-FFT intermediate, which is smaller).
    float* Fr = (float*)d_ws;
    float* Fi = Fr + (size_t)IMG_N * IMG_N;
    float* Tr = Fi + (size_t)IMG_N * IMG_N;
    float* Ti = Tr + (size_t)OUT_TOTAL;

    const int BLK = 256;
    const size_t colLds1024 = (size_t)4 * 16 * (1024 + 4) * sizeof(float); // ~257KB (fits 320KB WGP LDS)
    const size_t colLds512  = (size_t)4 * 16 * (512 + 4) * sizeof(float);
    const size_t colLds256  = (size_t)4 * 16 * (256 + 4) * sizeof(float);

    hipFuncSetAttribute((const void*)k_fwd_cols,
                        hipFuncAttributeMaxDynamicSharedMemorySize, (int)colLds1024);
    hipFuncSetAttribute((const void*)k_band_cols,
                        hipFuncAttributeMaxDynamicSharedMemorySize, (int)colLds1024);

    // Forward FFT2: rows -> (Tr,Ti), cols -> (Fr,Fi)
    k_fwd_rows<<<dim3(IMG_N), dim3(BLK), 4 * IMG_N * sizeof(float), stream>>>(image, Tr, Ti);
    k_fwd_cols<<<dim3(IMG_N / 16), dim3(BLK), colLds1024, stream>>>(Tr, Ti, Fr, Fi);

    // Groups: (d, winStart, nBands, nf, outOffset)
    // g0: d=4, wins 0..6,   nf=256,  off 0
    // g1: d=2, wins 7..18,  nf=512,  off 458752
    // g2: d=1, wins 19..42, nf=1024, off 3604480
    // Scale per group: 1 / (d * nf * nf)   [= d / (d^2 * nf^2)]

    // Group 0
    k_band_cols<<<dim3(256 / 16, 7), dim3(BLK), colLds256, stream>>>(
        Fr, Fi, windows, Tr, Ti, 256, 4);
    k_band_rows<<<dim3(256, 7), dim3(BLK), 4 * 256 * sizeof(float), stream>>>(
        Tr, Ti, out, 256, 1.0f / (4.0f * 256.0f * 256.0f));

    // Group 1
    k_band_cols<<<dim3(512 / 16, 12), dim3(BLK), colLds512, stream>>>(
        Fr, Fi, windows + (size_t)7 * IMG_N * IMG_N,
        Tr + (size_t)458752, Ti + (size_t)458752, 512, 2);
    k_band_rows<<<dim3(512, 12), dim3(BLK), 4 * 512 * sizeof(float), stream>>>(
        Tr + (size_t)458752, Ti + (size_t)458752, out + (size_t)458752,
        512, 1.0f / (2.0f * 512.0f * 512.0f));

    // Group 2
    k_band_cols<<<dim3(1024 / 16, 24), dim3(BLK), colLds1024, stream>>>(
        Fr, Fi, windows + (size_t)19 * IMG_N * IMG_N,
        Tr + (size_t)3604480, Ti + (size_t)3604480, 1024, 1);
    k_band_rows<<<dim3(1024, 24), dim3(BLK), 4 * 1024 * sizeof(float), stream>>>(
        Tr + (size_t)3604480, Ti + (size_t)3604480, out + (size_t)3604480,
        1024, 1.0f / (1024.0f * 1024.0f));
}